// TimesNet_4492535791824
// MI455X (gfx1250) — compile-verified
//
#include <hip/hip_runtime.h>
#include <stdint.h>

// Problem constants (from the reference): features/mask [BN, C, T] fp32, K output bins.
#define T_LEN 4096
#define C_CH  32
#define K_OUT 256

typedef __attribute__((ext_vector_type(4))) unsigned int u32x4;
typedef __attribute__((ext_vector_type(8))) unsigned int u32x8;

// Issue one TDM descriptor copying a contiguous row of T_LEN fp32 from global
// memory into LDS at byte offset lds_byte_off. Descriptor layout per CDNA5 ISA
// §8.3 (group 0) and §8.4 (group 1); 2-operand asm form => VADDR2/VADDR3 = NULL
// (tensor rank <= 2). Wave-level op: EXEC ignored, tracked by TENSORcnt.
// th:TH_LOAD_NT — rows are consumed exactly once (into LDS), so keep the
// 512 MiB stream from thrashing the 192 MB L2.
__device__ __forceinline__ void tdm_load_row_to_lds(const float* gsrc,
                                                    unsigned lds_byte_off) {
  unsigned long long ga = (unsigned long long)(uintptr_t)gsrc;
  u32x4 g0;
  g0[0] = 1u;                                      // count=1 (valid user D#), no gather
  g0[1] = lds_byte_off;                            // lds_addr (bytes)
  g0[2] = (unsigned)ga;                            // global_addr[31:0]
  g0[3] = ((unsigned)(ga >> 32) & 0x01FFFFFFu)     // global_addr[56:32]
          | (2u << 30);                            // type = 2 ("image")
  u32x8 g1;
  g1[0] = 2u << 16;                                // workgroup_mask=0, data_size=2 (4 B)
  g1[1] = (unsigned)T_LEN << 16;                   // tensor_dim0[15:0] = 4096
  g1[2] = 1u << 16;                                // tensor_dim0[31:16]=0 | tensor_dim1=1
  g1[3] = (unsigned)T_LEN << 16;                   // tensor_dim1[31:16]=0 | tile_dim0=4096
  g1[4] = 1u;                                      // tile_dim1=1, tile_dim2=0
  g1[5] = (unsigned)T_LEN;                         // tensor_dim0_stride[31:0] = 4096
  g1[6] = 0u;                                      // stride hi / dim1_stride lo
  g1[7] = 0u;                                      // dim1_stride hi
  asm volatile("tensor_load_to_lds %0, %1 th:TH_LOAD_NT"
               :: "s"(g0), "s"(g1) : "memory");
}

// One block per (bn, c) row-pair. TDM-stage both 16 KB rows into LDS, then each
// of the 256 threads reduces its adaptive-pool bin for features and mask.
// All bins lie inside the trailing valid window [T-L, T), so no masking needed:
//   si = off + floor(L*k/K) >= off,  ei = off + ceil(L*(k+1)/K) <= off + L = T,
//   cnt = ei - si >= 1 always.
__global__ __launch_bounds__(K_OUT) void adaptive_tail_pool_kernel(
    const float* __restrict__ feats, const float* __restrict__ mask,
    const int* __restrict__ vlen, float* __restrict__ out, int BN) {
  __shared__ float srow[2][T_LEN];                 // 32 KB of the 320 KB WGP LDS

  const int bc = (int)blockIdx.x;                  // bn * C + c
  const int bn = bc / C_CH;
  int L = vlen[bn];
  L = (L < 1) ? 1 : (L > T_LEN ? T_LEN : L);
  const int off = T_LEN - L;

  const float* frow = feats + (size_t)bc * T_LEN;
  const float* mrow = mask  + (size_t)bc * T_LEN;

  // Wave 0 issues both DMAs and waits for TENSORcnt==0; everyone else parks at
  // the barrier. TDM writes LDS directly, so post-barrier the data is visible.
  if (threadIdx.x < 32u) {
    // Low 32 bits of a generic LDS pointer are the workgroup-relative LDS byte
    // offset (ISA §10.2 aperture mapping).
    unsigned lds0 = (unsigned)(uintptr_t)&srow[0][0];
    unsigned lds1 = (unsigned)(uintptr_t)&srow[1][0];
    tdm_load_row_to_lds(frow, lds0);
    tdm_load_row_to_lds(mrow, lds1);
#if __has_builtin(__builtin_amdgcn_s_wait_tensorcnt)
    __builtin_amdgcn_s_wait_tensorcnt(0);
#else
    asm volatile("s_wait_tensorcnt 0x0" ::: "memory");
#endif
  }
  __syncthreads();

  // Bin k covers [off + floor(L*k/K), off + ceil(L*(k+1)/K)); width <= 17.
  const int k   = (int)threadIdx.x;
  const int s   = off + (L * k) / K_OUT;
  const int e   = off + (L * (k + 1) + K_OUT - 1) / K_OUT;
  const int cnt = e - s;                           // always >= 1

  float fs = 0.0f, ms = 0.0f;
  for (int i = s; i < e; ++i) {
    fs += srow[0][i];
    ms += srow[1][i];
  }
  const float invc = 1.0f / (float)cnt;

  // Write-once outputs: non-temporal stores (th:TH_STORE_NT) to avoid cache
  // pollution; stores are fully coalesced (thread k -> element k of the row).
  const size_t o = (size_t)bc * K_OUT + (size_t)k;
  __builtin_nontemporal_store(fs * invc, out + o);                           // pooled_feats
  __builtin_nontemporal_store(ms * invc, out + (size_t)BN * C_CH * K_OUT + o); // pooled_mask
}

extern "C" void kernel_launch(void* const* d_in, const int* in_sizes, int n_in,
                              void* d_out, int out_size, void* d_ws, size_t ws_size,
                              hipStream_t stream) {
  (void)n_in; (void)out_size; (void)d_ws; (void)ws_size;
  const float* feats = (const float*)d_in[0];   // [BN, C, T] fp32
  const float* mask  = (const float*)d_in[1];   // [BN, C, T] fp32
  const int*   vlen  = (const int*)d_in[2];     // [BN] (int32 on device; JAX x64 off)
  // d_in[3] = target_len scalar; K fixed at 256 to match the reference dispatch.
  float* out = (float*)d_out;                   // [2, BN, C, K] fp32, concatenated

  const int BN = in_sizes[0] / (C_CH * T_LEN);  // 512
  dim3 grid((unsigned)(BN * C_CH));             // 16384 blocks
  adaptive_tail_pool_kernel<<<grid, K_OUT, 0, stream>>>(feats, mask, vlen, out, BN);
}